// SubQmixer_50491635532296
// MI455X (gfx1250) — compile-verified
//
#include <hip/hip_runtime.h>
#include <hip/hip_bf16.h>
#include <cstdint>

#define DIM 64
#define NODE_ALLY 0
#define TARGET_ASSIGNMENT 0

typedef __attribute__((ext_vector_type(2))) float v2f;
typedef __attribute__((ext_vector_type(8))) float v8f;

static __device__ __forceinline__ v8f wmma_f32_k4(v2f a, v2f b, v8f c) {
  // D = A(16x4,f32) * B(4x16,f32) + C(16x16,f32)
  return __builtin_amdgcn_wmma_f32_16x16x4_f32(false, a, false, b, (short)0, c,
                                               false, false);
}

static __device__ __forceinline__ void fatomic_add(float* p, float v) {
  unsafeAtomicAdd(p, v);  // -> global_atomic_add_f32
}

// ---------------------------------------------------------------- zero fills
__global__ void zero_f32(float* __restrict__ p, size_t n) {
  size_t i = (size_t)blockIdx.x * blockDim.x + threadIdx.x;
  if (i < n) p[i] = 0.0f;
}
__global__ void zero_u8(unsigned char* __restrict__ p, size_t n) {
  size_t i = (size_t)blockIdx.x * blockDim.x + threadIdx.x;
  if (i < n) p[i] = 0;
}

// --------------------------------------------------- stage 1: 8 GEMMs, WMMA
// Block = 256 threads = 8 waves; the block async-copies one 16x64 H tile into
// LDS (global_load_async_to_lds_b128), then each wave owns one 64x64 weight
// matrix and produces a full 16x64 output tile (4 col tiles x 16 K-steps).
__global__ void __launch_bounds__(256)
rgcn_stage1(const float* __restrict__ H,    // [N,64]
            const float* __restrict__ Wwr,  // [3,64,64]
            const float* __restrict__ Wws,  // [64,64]
            const float* __restrict__ Wvr,  // [3,64,64]
            const float* __restrict__ Wvs,  // [64,64]
            float* __restrict__ t_w,        // [3,N,64]
            float* __restrict__ t_v,        // [3,N,64]
            float* __restrict__ self_w,     // [N,64]
            float* __restrict__ self_v,     // [N,64]
            int N) {
  __shared__ float s_A[16 * DIM];

  const int tid  = threadIdx.x;
  const int wv   = tid >> 5;
  const int lane = tid & 31;
  const int half = lane >> 4;   // lanes 16-31 hold K+2 / M+8 per ISA layout
  const int l15  = lane & 15;
  const int m0   = blockIdx.x * 16;

  // ---- async stage of the 16x64 A tile: one B128 per thread -------------
  {
    int row = tid >> 4;          // 0..15
    int cg  = tid & 15;          // column group of 4 floats
    int ar = m0 + row; if (ar >= N) ar = N - 1;   // clamp tail tiles
    const float* gsrc = H + (size_t)ar * DIM + cg * 4;
    unsigned lds_off = (unsigned)(uintptr_t)(&s_A[0]) + (unsigned)(tid * 16);
    asm volatile("global_load_async_to_lds_b128 %0, %1, off"
                 :
                 : "v"(lds_off), "v"(gsrc)
                 : "memory");
  }
  asm volatile("s_wait_asynccnt 0x0" ::: "memory");
  __syncthreads();

  const float* W;
  float* Out;
  if (wv < 3)       { W = Wwr + wv * DIM * DIM;       Out = t_w + (size_t)wv * N * DIM; }
  else if (wv == 3) { W = Wws;                        Out = self_w; }
  else if (wv < 7)  { W = Wvr + (wv - 4) * DIM * DIM; Out = t_v + (size_t)(wv - 4) * N * DIM; }
  else              { W = Wvs;                        Out = self_v; }

  for (int nt = 0; nt < 4; ++nt) {
    const int n0 = nt * 16;
    v8f acc = {};
#pragma unroll
    for (int k0 = 0; k0 < DIM; k0 += 4) {
      v2f a = *(const v2f*)&s_A[l15 * DIM + k0 + 2 * half];  // ds_load_b64
      v2f b;
      b.x = W[(k0 + 2 * half + 0) * DIM + n0 + l15];
      b.y = W[(k0 + 2 * half + 1) * DIM + n0 + l15];
      acc = wmma_f32_k4(a, b, acc);
    }
#pragma unroll
    for (int v = 0; v < 8; ++v) {
      int row = m0 + half * 8 + v;
      if (row < N) Out[(size_t)row * DIM + n0 + l15] = acc[v];
    }
  }
}

// ------------------------------------------- edge gather/scatter (L2-bound)
// One thread per (edge, 4 features): b128 gathers + 4 FP atomics each.
__global__ void __launch_bounds__(256)
edge_scatter(const float* __restrict__ t_w, const float* __restrict__ t_v,
             const int* __restrict__ esrc, const int* __restrict__ edst,
             const int* __restrict__ etyp,
             float* __restrict__ agg_w, float* __restrict__ agg_v,
             float* __restrict__ degf, long E, int N) {
  long gid = (long)blockIdx.x * blockDim.x + threadIdx.x;
  if (gid >= E * (DIM / 4)) return;
  int e  = (int)(gid >> 4);
  int j4 = (int)(gid & 15) * 4;
  int s = esrc[e], d = edst[e], r = etyp[e];
  size_t si = ((size_t)r * N + s) * DIM + j4;
  size_t di = (size_t)d * DIM + j4;
  float4 tw = *(const float4*)(t_w + si);   // global_load_b128
  float4 tv = *(const float4*)(t_v + si);
  fatomic_add(&agg_w[di + 0], tw.x);
  fatomic_add(&agg_w[di + 1], tw.y);
  fatomic_add(&agg_w[di + 2], tw.z);
  fatomic_add(&agg_w[di + 3], tw.w);
  fatomic_add(&agg_v[di + 0], tv.x);
  fatomic_add(&agg_v[di + 1], tv.y);
  fatomic_add(&agg_v[di + 2], tv.z);
  fatomic_add(&agg_v[di + 3], tv.w);
  if (j4 == 0) fatomic_add(&degf[d], 1.0f);
}

// --------------------- FF: emb=relu(agg/deg+self); hid=relu(emb@W1+b1); dot W2
__global__ void __launch_bounds__(256)
rgcn_ff(const float* __restrict__ agg_w, const float* __restrict__ agg_v,
        const float* __restrict__ self_w, const float* __restrict__ self_v,
        const float* __restrict__ degf,
        const float* __restrict__ Wwf1, const float* __restrict__ bwf1,
        const float* __restrict__ Wwf2, const float* __restrict__ bwf2,
        const float* __restrict__ Wvf1, const float* __restrict__ bvf1,
        const float* __restrict__ Wvf2, const float* __restrict__ bvf2,
        float* __restrict__ w_out, float* __restrict__ v_out, int N) {
  __shared__ float s_emb_w[16 * DIM];
  __shared__ float s_emb_v[16 * DIM];
  __shared__ float s_hid_w[16 * DIM];
  __shared__ float s_hid_v[16 * DIM];
  __shared__ float s_red[256];

  const int tid = threadIdx.x;
  const int m0 = blockIdx.x * 16;

  // phase 1: embeddings into LDS (float4 per thread: 256 * 4 = 1024 elems)
  {
    int idx4 = tid * 4;
    int nl = idx4 >> 6, j = idx4 & 63;
    int n = m0 + nl;
    int nc = n < N ? n : N - 1;
    float dg = degf[nc]; dg = dg > 1.0f ? dg : 1.0f;
    size_t o = (size_t)nc * DIM + j;
    float4 aw = *(const float4*)(agg_w + o);
    float4 sw = *(const float4*)(self_w + o);
    float4 av = *(const float4*)(agg_v + o);
    float4 sv = *(const float4*)(self_v + o);
    float ew0 = aw.x / dg + sw.x, ew1 = aw.y / dg + sw.y;
    float ew2 = aw.z / dg + sw.z, ew3 = aw.w / dg + sw.w;
    float ev0 = av.x / dg + sv.x, ev1 = av.y / dg + sv.y;
    float ev2 = av.z / dg + sv.z, ev3 = av.w / dg + sv.w;
    s_emb_w[idx4 + 0] = ew0 > 0.0f ? ew0 : 0.0f;
    s_emb_w[idx4 + 1] = ew1 > 0.0f ? ew1 : 0.0f;
    s_emb_w[idx4 + 2] = ew2 > 0.0f ? ew2 : 0.0f;
    s_emb_w[idx4 + 3] = ew3 > 0.0f ? ew3 : 0.0f;
    s_emb_v[idx4 + 0] = ev0 > 0.0f ? ev0 : 0.0f;
    s_emb_v[idx4 + 1] = ev1 > 0.0f ? ev1 : 0.0f;
    s_emb_v[idx4 + 2] = ev2 > 0.0f ? ev2 : 0.0f;
    s_emb_v[idx4 + 3] = ev3 > 0.0f ? ev3 : 0.0f;
  }
  __syncthreads();

  // phase 2: hidden = relu(emb @ Wff1 + b1) via WMMA (waves 0-3: w, 4-7: v)
  {
    const int wv = tid >> 5;
    const int lane = tid & 31;
    const int half = lane >> 4;
    const int l15 = lane & 15;
    const int br = wv >> 2;
    const int n0 = (wv & 3) * 16;
    const float* emb = br ? s_emb_v : s_emb_w;
    const float* W1  = br ? Wvf1 : Wwf1;
    const float* b1  = br ? bvf1 : bwf1;
    float* hid = br ? s_hid_v : s_hid_w;
    v8f acc = {};
#pragma unroll
    for (int k0 = 0; k0 < DIM; k0 += 4) {
      v2f a = *(const v2f*)&emb[l15 * DIM + k0 + 2 * half];  // ds_load_b64
      v2f b;
      b.x = W1[(k0 + 2 * half + 0) * DIM + n0 + l15];
      b.y = W1[(k0 + 2 * half + 1) * DIM + n0 + l15];
      acc = wmma_f32_k4(a, b, acc);
    }
    float bb = b1[n0 + l15];
#pragma unroll
    for (int v = 0; v < 8; ++v) {
      float h = acc[v] + bb;
      hid[(half * 8 + v) * DIM + n0 + l15] = h > 0.0f ? h : 0.0f;
    }
  }
  __syncthreads();

  // phase 3: per-node dot with Wff2 (8 threads per (node,branch))
  {
    const int br = tid >> 7;
    const int nl = (tid >> 3) & 15;
    const int j0 = (tid & 7) * 8;
    const float* hid = br ? s_hid_v : s_hid_w;
    const float* W2 = br ? Wvf2 : Wwf2;
    float p = 0.0f;
#pragma unroll
    for (int j = 0; j < 8; ++j) p += hid[nl * DIM + j0 + j] * W2[j0 + j];
    s_red[tid] = p;
  }
  __syncthreads();
  if ((tid & 7) < 4) s_red[tid] += s_red[tid + 4];
  __syncthreads();
  if ((tid & 7) < 2) s_red[tid] += s_red[tid + 2];
  __syncthreads();
  if ((tid & 7) == 0) {
    float val = s_red[tid] + s_red[tid + 1];
    const int br = tid >> 7;
    const int nl = (tid >> 3) & 15;
    int n = m0 + nl;
    if (n < N) {
      if (br == 0) w_out[n] = fabsf(val + bwf2[0]);
      else         v_out[n] = val + bvf2[0];
    }
  }
}

// ------------------------------- prefix scan of is_ally to build tmask scatter
__global__ void scan_block(const int* __restrict__ ntyp, int* __restrict__ bsum,
                           int N) {
  __shared__ int s[256];
  int tid = threadIdx.x;
  int n = blockIdx.x * 256 + tid;
  s[tid] = (n < N && ntyp[n] == NODE_ALLY) ? 1 : 0;
  __syncthreads();
  for (int off = 128; off > 0; off >>= 1) {
    if (tid < off) s[tid] += s[tid + off];
    __syncthreads();
  }
  if (tid == 0) bsum[blockIdx.x] = s[0];
}

__global__ void exscan_serial(int* __restrict__ bsum, int nb) {
  if (blockIdx.x == 0 && threadIdx.x == 0) {
    int acc = 0;
    for (int i = 0; i < nb; ++i) { int t = bsum[i]; bsum[i] = acc; acc += t; }
  }
}

__global__ void scan_apply(const int* __restrict__ ntyp,
                           const int* __restrict__ assign,
                           const int* __restrict__ bsum,
                           unsigned char* __restrict__ tmask, int N) {
  __shared__ int s[256];
  int tid = threadIdx.x;
  int n = blockIdx.x * 256 + tid;
  int a = (n < N && ntyp[n] == NODE_ALLY) ? 1 : 0;
  s[tid] = a;
  __syncthreads();
  for (int off = 1; off < 256; off <<= 1) {  // Hillis-Steele inclusive scan
    int val = (tid >= off) ? s[tid - off] : 0;
    __syncthreads();
    s[tid] += val;
    __syncthreads();
  }
  if (a && n < N && assign[n] == TARGET_ASSIGNMENT) {
    int rank = bsum[blockIdx.x] + s[tid] - 1;  // ally-relative index, < N
    tmask[rank] = 1;
  }
}

// --------------------------------------------------- final per-graph reduction
__global__ void finalize(const unsigned char* __restrict__ tmask,
                         const float* __restrict__ w, const float* __restrict__ v,
                         const float* __restrict__ qs, const int* __restrict__ gid,
                         float* __restrict__ out, int N) {
  int n = blockIdx.x * blockDim.x + threadIdx.x;
  if (n < N && tmask[n]) fatomic_add(&out[gid[n]], w[n] * qs[n] + v[n]);
}

extern "C" void kernel_launch(void* const* d_in, const int* in_sizes, int n_in,
                              void* d_out, int out_size, void* d_ws,
                              size_t ws_size, hipStream_t stream) {
  const float* H    = (const float*)d_in[0];
  const float* qs   = (const float*)d_in[1];
  const float* Wwr  = (const float*)d_in[2];
  const float* Wws  = (const float*)d_in[3];
  const float* Wwf1 = (const float*)d_in[4];
  const float* bwf1 = (const float*)d_in[5];
  const float* Wwf2 = (const float*)d_in[6];
  const float* bwf2 = (const float*)d_in[7];
  const float* Wvr  = (const float*)d_in[8];
  const float* Wvs  = (const float*)d_in[9];
  const float* Wvf1 = (const float*)d_in[10];
  const float* bvf1 = (const float*)d_in[11];
  const float* Wvf2 = (const float*)d_in[12];
  const float* bvf2 = (const float*)d_in[13];
  const int* esrc   = (const int*)d_in[14];
  const int* edst   = (const int*)d_in[15];
  const int* etyp   = (const int*)d_in[16];
  const int* ntyp   = (const int*)d_in[17];
  const int* assign = (const int*)d_in[18];
  const int* gids   = (const int*)d_in[19];

  const int N = in_sizes[1];
  const long E = in_sizes[14];
  const int B = out_size;
  float* out = (float*)d_out;

  // workspace carve-up (floats)
  float* ws = (float*)d_ws;
  size_t o = 0;
  float* t_w    = ws + o; o += (size_t)3 * N * DIM;
  float* t_v    = ws + o; o += (size_t)3 * N * DIM;
  float* self_w = ws + o; o += (size_t)N * DIM;
  float* self_v = ws + o; o += (size_t)N * DIM;
  float* agg_w  = ws + o; o += (size_t)N * DIM;   // agg_w/agg_v/deg contiguous
  float* agg_v  = ws + o; o += (size_t)N * DIM;
  float* degf   = ws + o; o += (size_t)N;
  float* w_arr  = ws + o; o += (size_t)N;
  float* v_arr  = ws + o; o += (size_t)N;
  int nb = (N + 255) / 256;
  int* bsum = (int*)(ws + o); o += (size_t)nb;
  unsigned char* tmask = (unsigned char*)(ws + o);

  // 1. zero accumulators / mask / output
  size_t zc = (size_t)2 * N * DIM + N;  // agg_w + agg_v + deg
  zero_f32<<<(int)((zc + 255) / 256), 256, 0, stream>>>(agg_w, zc);
  zero_u8<<<(N + 255) / 256, 256, 0, stream>>>(tmask, (size_t)N);
  zero_f32<<<(B + 255) / 256, 256, 0, stream>>>(out, (size_t)B);

  // 2. relation/self GEMMs (WMMA, async LDS staging of A tiles)
  int mtiles = (N + 15) / 16;
  rgcn_stage1<<<mtiles, 256, 0, stream>>>(H, Wwr, Wws, Wvr, Wvs,
                                          t_w, t_v, self_w, self_v, N);

  // 3. edge gather/scatter with b128 gathers + FP atomics
  long tot = E * (DIM / 4);
  edge_scatter<<<(int)((tot + 255) / 256), 256, 0, stream>>>(
      t_w, t_v, esrc, edst, etyp, agg_w, agg_v, degf, E, N);

  // 4. mean-agg + self + relu, FF head (WMMA), per-node scalars
  rgcn_ff<<<mtiles, 256, 0, stream>>>(agg_w, agg_v, self_w, self_v, degf,
                                      Wwf1, bwf1, Wwf2, bwf2,
                                      Wvf1, bvf1, Wvf2, bvf2,
                                      w_arr, v_arr, N);

  // 5. ally-rank prefix scan -> tmask scatter (reference semantics)
  scan_block<<<nb, 256, 0, stream>>>(ntyp, bsum, N);
  exscan_serial<<<1, 1, 0, stream>>>(bsum, nb);
  scan_apply<<<nb, 256, 0, stream>>>(ntyp, assign, bsum, tmask, N);

  // 6. masked per-graph reduction
  finalize<<<(N + 255) / 256, 256, 0, stream>>>(tmask, w_arr, v_arr, qs, gids,
                                                out, N);
}